// PrototypeLayer_1108101562410
// MI455X (gfx1250) — compile-verified
//
#include <hip/hip_runtime.h>
#include <hip/hip_bf16.h>

typedef __attribute__((ext_vector_type(2))) float v2f;
typedef __attribute__((ext_vector_type(8))) float v8f;

// Problem constants (from reference)
constexpr int BB = 256;   // batch of input subspaces
constexpr int PP = 200;   // prototypes
constexpr int DD = 256;   // ambient dim
constexpr int KK = 8;     // subspace dim
constexpr int WAVES_PER_BLOCK = 8;
constexpr int NPAIR = BB * PP;                  // 51200
constexpr int NBLOCK = NPAIR / WAVES_PER_BLOCK; // 6400 exact

// Per-wave LDS layout (floats)
constexpr int L_PROJ  = 0;    // 8x8 proj
constexpr int L_G     = 64;   // 8x8 Gram
constexpr int L_V     = 128;  // 8x8 eigenvectors
constexpr int L_LAM   = 192;  // 8 eigenvalues
constexpr int L_STAGE = 200;  // 16x8 store staging (800B offset -> 16B aligned)
constexpr int L_WAVE  = 336;  // per-wave stride (1344B, 16B aligned)

__global__ __launch_bounds__(256) void
grassmann_proto_kernel(const float* __restrict__ X,   // (B,D,K)
                       const float* __restrict__ W,   // (P,D,K)
                       const float* __restrict__ R,   // (1,K)
                       float* __restrict__ dist,      // (B,P)
                       float* __restrict__ Qw)        // (B,P,D,K)
{
    __shared__ float lds[WAVES_PER_BLOCK * L_WAVE];

    const int tid  = threadIdx.x;
    const int wave = tid >> 5;
    const int lane = tid & 31;
    float* wl = &lds[wave * L_WAVE];

    const int pair = blockIdx.x * WAVES_PER_BLOCK + wave;
    const int b = pair / PP;
    const int p = pair % PP;

    const float* Xb = X + (size_t)b * (DD * KK);
    const float* Wp = W + (size_t)p * (DD * KK);

    const int half = lane >> 4;   // selects which K-pair of the chunk this lane holds
    const int l16  = lane & 15;   // A: M index / B,C,D: N index
    const int idx8 = l16 & 7;     // wrapped always-valid column index (no masking
                                  // needed: result rows/cols 8..15 are never read)
    const bool kv  = (l16 < KK);

    // ---------------- proj = X^T W  (16x16 padded, K-loop over D in steps of 4) ----
    // A[m][c] = X[d0+2*half+c][m], B[c][n] = W[d0+2*half+c][n]  (c = 0,1 per VGPR)
    // Lanes with l16>=8 load wrapped data; the garbage lands only in unused
    // rows/cols of C. All loads are UNCONDITIONAL -> no exec-mask branches.
    v8f acc = {};
    for (int d0 = 0; d0 < DD; d0 += 4) {
        const int ka = d0 + 2 * half;
        v2f a, bf;
        a.x  = Xb[(ka + 0) * KK + idx8];
        a.y  = Xb[(ka + 1) * KK + idx8];
        bf.x = Wp[(ka + 0) * KK + idx8];
        bf.y = Wp[(ka + 1) * KK + idx8];
        acc = __builtin_amdgcn_wmma_f32_16x16x4_f32(false, a, false, bf,
                                                    (short)0, acc, false, false);
    }

    // Spill proj (rows 0..7 live in half==0 lanes, VGPR j = row j) to LDS
    if (half == 0 && kv) {
        #pragma unroll
        for (int j = 0; j < 8; ++j) wl[L_PROJ + j * 8 + l16] = acc[j];
    }
    __builtin_amdgcn_wave_barrier();

    // ---------------- lane 0: G = proj^T proj, Jacobi eigensolve, sort, distance ----
    if (lane == 0) {
        for (int i = 0; i < 8; ++i)
            for (int j = 0; j < 8; ++j) {
                float s = 0.0f;
                for (int m = 0; m < 8; ++m)
                    s += wl[L_PROJ + m * 8 + i] * wl[L_PROJ + m * 8 + j];
                wl[L_G + i * 8 + j] = s;
            }
        for (int i = 0; i < 8; ++i)
            for (int j = 0; j < 8; ++j)
                wl[L_V + i * 8 + j] = (i == j) ? 1.0f : 0.0f;

        // cyclic Jacobi sweeps on symmetric 8x8
        for (int sweep = 0; sweep < 6; ++sweep) {
            for (int pi = 0; pi < 7; ++pi) {
                for (int qi = pi + 1; qi < 8; ++qi) {
                    float apq = wl[L_G + pi * 8 + qi];
                    if (__builtin_fabsf(apq) < 1e-12f) continue;
                    float app = wl[L_G + pi * 8 + pi];
                    float aqq = wl[L_G + qi * 8 + qi];
                    float tau = (aqq - app) / (2.0f * apq);
                    float t = ((tau >= 0.0f) ? 1.0f : -1.0f) /
                              (__builtin_fabsf(tau) + __builtin_sqrtf(1.0f + tau * tau));
                    float c = 1.0f / __builtin_sqrtf(1.0f + t * t);
                    float s = t * c;
                    for (int m = 0; m < 8; ++m) {   // column rotation
                        float gp = wl[L_G + m * 8 + pi], gq = wl[L_G + m * 8 + qi];
                        wl[L_G + m * 8 + pi] = c * gp - s * gq;
                        wl[L_G + m * 8 + qi] = s * gp + c * gq;
                    }
                    for (int m = 0; m < 8; ++m) {   // row rotation
                        float gp = wl[L_G + pi * 8 + m], gq = wl[L_G + qi * 8 + m];
                        wl[L_G + pi * 8 + m] = c * gp - s * gq;
                        wl[L_G + qi * 8 + m] = s * gp + c * gq;
                    }
                    for (int m = 0; m < 8; ++m) {   // eigenvector columns
                        float vp = wl[L_V + m * 8 + pi], vq = wl[L_V + m * 8 + qi];
                        wl[L_V + m * 8 + pi] = c * vp - s * vq;
                        wl[L_V + m * 8 + qi] = s * vp + c * vq;
                    }
                }
            }
        }
        for (int k = 0; k < 8; ++k) wl[L_LAM + k] = wl[L_G + k * 8 + k];
        // sort eigenvalues descending (match SVD ordering), permute V columns
        for (int i = 0; i < 8; ++i) {
            int best = i;
            for (int j = i + 1; j < 8; ++j)
                if (wl[L_LAM + j] > wl[L_LAM + best]) best = j;
            if (best != i) {
                float tl = wl[L_LAM + i]; wl[L_LAM + i] = wl[L_LAM + best]; wl[L_LAM + best] = tl;
                for (int m = 0; m < 8; ++m) {
                    float tv = wl[L_V + m * 8 + i];
                    wl[L_V + m * 8 + i] = wl[L_V + m * 8 + best];
                    wl[L_V + m * 8 + best] = tv;
                }
            }
        }
        // d = 1 - sum_k r_k * lambda_k   (lambda = s^2)
        float dsum = 0.0f;
        for (int k = 0; k < 8; ++k) dsum += R[k] * wl[L_LAM + k];
        dist[b * PP + p] = 1.0f - dsum;
    }
    __builtin_amdgcn_wave_barrier();

    // ---------------- Qw = W_p * V  (16 row-tiles of 16, K-loop 2x4, N=8 padded) ----
    float* stage = wl + L_STAGE;
    float* Qwp   = Qw + (size_t)pair * (DD * KK);
    const int dofs = 2 * half;

    for (int t = 0; t < 16; ++t) {
        const int m0 = t * 16;
        v8f dfrag = {};
        #pragma unroll
        for (int kk = 0; kk < KK; kk += 4) {
            const int row = m0 + l16;                 // A: M = W row
            // A pair is contiguous & 8B aligned -> single b64 load
            const float2 av = *(const float2*)&Wp[row * KK + kk + dofs];
            v2f a, bf;
            a.x  = av.x;
            a.y  = av.y;
            // B from LDS; wrapped index for n>=8 lanes (unused output columns)
            bf.x = wl[L_V + (kk + dofs + 0) * 8 + idx8];
            bf.y = wl[L_V + (kk + dofs + 1) * 8 + idx8];
            dfrag = __builtin_amdgcn_wmma_f32_16x16x4_f32(false, a, false, bf,
                                                          (short)0, dfrag, false, false);
        }
        // stage tile (16 rows x 8 cols) to LDS in memory order
        if (kv) {
            #pragma unroll
            for (int j = 0; j < 8; ++j)
                stage[(8 * half + j) * 8 + l16] = dfrag[j];
        }
        __builtin_amdgcn_wave_barrier();
        // coalesced 128-bit store: 32 lanes x float4 = whole 512B tile
        const float4 val = *(const float4*)&stage[lane * 4];
        *(float4*)&Qwp[m0 * KK + lane * 4] = val;
        __builtin_amdgcn_wave_barrier();
    }
}

extern "C" void kernel_launch(void* const* d_in, const int* in_sizes, int n_in,
                              void* d_out, int out_size, void* d_ws, size_t ws_size,
                              hipStream_t stream) {
    const float* X = (const float*)d_in[0];   // xs_subspace (B,D,K)
    const float* W = (const float*)d_in[1];   // xprotos     (P,D,K)
    const float* R = (const float*)d_in[2];   // relevances  (1,K)
    float* dist = (float*)d_out;              // (B,P)
    float* Qw   = (float*)d_out + (size_t)BB * PP;  // (B,P,D,K)
    grassmann_proto_kernel<<<dim3(NBLOCK), dim3(256), 0, stream>>>(X, W, R, dist, Qw);
}